// H2GCN_86947317941139
// MI455X (gfx1250) — compile-verified
//
#include <hip/hip_runtime.h>

#define LD 960  // concat width: 64 (h) + 128 (c0) + 256 (c1) + 512 (c2)

typedef float v2f __attribute__((ext_vector_type(2)));
typedef float v8f __attribute__((ext_vector_type(8)));

// ---------------------------------------------------------------- zero
__global__ void zero_f32(float* __restrict__ p, long long cnt) {
  long long i = blockIdx.x * (long long)blockDim.x + threadIdx.x;
  const long long s = (long long)gridDim.x * blockDim.x;
  for (; i < cnt; i += s) p[i] = 0.0f;
}

// ------------------------------------------------- embed: relu(xW+b) -> J[:,0:64]
// one wave per 16-row stripe; 4 column tiles kept in registers so each A
// fragment is loaded once and feeds 4 WMMAs (x streamed from HBM exactly once)
__global__ void embed_gemm_wmma(const float* __restrict__ A,
                                const float* __restrict__ W,   // [256,64]
                                const float* __restrict__ bias,
                                float* __restrict__ J, int n) {
  const int lane = threadIdx.x & 31;
  const int wave = threadIdx.x >> 5;
  const int rowTile = blockIdx.x * (blockDim.x >> 5) + wave;
  const int rowTiles = (n + 15) >> 4;
  if (rowTile >= rowTiles) return;             // wave-uniform
  const int hi  = lane >> 4;                   // 0: lanes 0-15, 1: lanes 16-31
  const int l15 = lane & 15;
  const int mrow = rowTile * 16 + l15;
  const int m  = mrow < n ? mrow : n - 1;      // clamp loads, EXEC stays full
  const float* __restrict__ arow = A + (size_t)m * 256;

  v8f acc0 = {}, acc1 = {}, acc2 = {}, acc3 = {};
  for (int k0 = 0; k0 < 256; k0 += 4) {
    const int ka = k0 + 2 * hi;
    v2f a;
    a.x = arow[ka];
    a.y = arow[ka + 1];
    const float* __restrict__ w0 = W + (size_t)ka * 64 + l15;        // row ka
    const float* __restrict__ w1 = W + (size_t)(ka + 1) * 64 + l15;  // row ka+1
    v2f b0, b1, b2, b3;
    b0.x = w0[0];  b0.y = w1[0];
    b1.x = w0[16]; b1.y = w1[16];
    b2.x = w0[32]; b2.y = w1[32];
    b3.x = w0[48]; b3.y = w1[48];
    acc0 = __builtin_amdgcn_wmma_f32_16x16x4_f32(false, a, false, b0, (short)0, acc0, false, false);
    acc1 = __builtin_amdgcn_wmma_f32_16x16x4_f32(false, a, false, b1, (short)0, acc1, false, false);
    acc2 = __builtin_amdgcn_wmma_f32_16x16x4_f32(false, a, false, b2, (short)0, acc2, false, false);
    acc3 = __builtin_amdgcn_wmma_f32_16x16x4_f32(false, a, false, b3, (short)0, acc3, false, false);
  }
  const float bv0 = bias[l15];
  const float bv1 = bias[l15 + 16];
  const float bv2 = bias[l15 + 32];
  const float bv3 = bias[l15 + 48];
#pragma unroll
  for (int r = 0; r < 8; ++r) {
    const int orow = rowTile * 16 + r + 8 * hi;  // D: VGPR r -> row r + 8*hi
    if (orow < n) {
      float* __restrict__ dst = J + (size_t)orow * LD;
      float v0 = acc0[r] + bv0, v1 = acc1[r] + bv1;
      float v2 = acc2[r] + bv2, v3 = acc3[r] + bv3;
      dst[l15]      = v0 > 0.0f ? v0 : 0.0f;
      dst[l15 + 16] = v1 > 0.0f ? v1 : 0.0f;
      dst[l15 + 32] = v2 > 0.0f ? v2 : 0.0f;
      dst[l15 + 48] = v3 > 0.0f ? v3 : 0.0f;
    }
  }
}

// ------------------------------------------------- head: out = J @ W_head + b
// 2 column tiles per wave -> the 192MB J matrix is streamed from HBM once
__global__ void head_gemm_wmma(const float* __restrict__ J,
                               const float* __restrict__ W,   // [960,32]
                               const float* __restrict__ bias,
                               float* __restrict__ out, int n) {
  const int lane = threadIdx.x & 31;
  const int wave = threadIdx.x >> 5;
  const int rowTile = blockIdx.x * (blockDim.x >> 5) + wave;
  const int rowTiles = (n + 15) >> 4;
  if (rowTile >= rowTiles) return;             // wave-uniform
  const int hi  = lane >> 4;
  const int l15 = lane & 15;
  const int mrow = rowTile * 16 + l15;
  const int m  = mrow < n ? mrow : n - 1;
  const float* __restrict__ jrow = J + (size_t)m * LD;

  v8f acc0 = {}, acc1 = {};
  for (int k0 = 0; k0 < LD; k0 += 4) {
    const int ka = k0 + 2 * hi;
    v2f a;
    a.x = jrow[ka];
    a.y = jrow[ka + 1];
    const float* __restrict__ w0 = W + (size_t)ka * 32 + l15;
    const float* __restrict__ w1 = W + (size_t)(ka + 1) * 32 + l15;
    v2f b0, b1;
    b0.x = w0[0];  b0.y = w1[0];
    b1.x = w0[16]; b1.y = w1[16];
    acc0 = __builtin_amdgcn_wmma_f32_16x16x4_f32(false, a, false, b0, (short)0, acc0, false, false);
    acc1 = __builtin_amdgcn_wmma_f32_16x16x4_f32(false, a, false, b1, (short)0, acc1, false, false);
  }
  const float bv0 = bias[l15];
  const float bv1 = bias[l15 + 16];
#pragma unroll
  for (int r = 0; r < 8; ++r) {
    const int orow = rowTile * 16 + r + 8 * hi;
    if (orow < n) {
      out[(size_t)orow * 32 + l15]      = acc0[r] + bv0;
      out[(size_t)orow * 32 + l15 + 16] = acc1[r] + bv1;
    }
  }
}

// ------------------------------------------------- SPMM scatter: out[row] += val * in[col]
// in/out point inside J (ld = LD); d4 = feature_dim / 4; float4 gather + 4 fp32 atomics
__global__ void spmm_scatter(const int* __restrict__ row, const int* __restrict__ col,
                             const float* __restrict__ val, int E,
                             const float* __restrict__ in, float* __restrict__ out,
                             int d4) {
  long long tid = blockIdx.x * (long long)blockDim.x + threadIdx.x;
  const long long total  = (long long)E * d4;
  const long long stride = (long long)gridDim.x * blockDim.x;
  for (; tid < total; tid += stride) {
    const int e = (int)(tid / d4);
    const int c = (int)(tid % d4);
    const int r  = row[e];
    const int cc = col[e];
    const float v = val[e];
    const float4 x = *((const float4*)(in + (size_t)cc * LD) + c);  // global_load_b128
    float* dst = out + (size_t)r * LD + c * 4;
    atomicAdd(dst + 0, v * x.x);
    atomicAdd(dst + 1, v * x.y);
    atomicAdd(dst + 2, v * x.z);
    atomicAdd(dst + 3, v * x.w);
  }
}

// ------------------------------------------------- BN partial sums (coalesced)
// block = d threads (one per column); each block covers a chunk of rows;
// per-thread local accumulation, then one atomicAdd per (block, column)
__global__ void bn_partial(const float* __restrict__ J, int n, int off, int d,
                           float* __restrict__ sum, float* __restrict__ sumsq) {
  const int c = threadIdx.x;                   // column (blockDim.x == d)
  float s = 0.0f, sq = 0.0f;
  for (int r = blockIdx.x; r < n; r += gridDim.x) {
    const float v = J[(size_t)r * LD + off + c];   // coalesced row segment
    s += v; sq += v * v;
  }
  atomicAdd(&sum[c], s);
  atomicAdd(&sumsq[c], sq);
}

__global__ void bn_finalize(const float* __restrict__ sum, const float* __restrict__ sumsq,
                            int n, int d,
                            float* __restrict__ mean, float* __restrict__ inv) {
  const int c = blockIdx.x * blockDim.x + threadIdx.x;
  if (c < d) {
    const float mn  = sum[c] / (float)n;
    const float var = sumsq[c] / (float)n - mn * mn;
    mean[c] = mn;
    inv[c]  = rsqrtf(var + 1e-5f);
  }
}

// ------------------------------------------------- BN apply (in place, coalesced)
__global__ void bn_apply(float* __restrict__ J, int n, int off, int d,
                         const float* __restrict__ mean, const float* __restrict__ inv,
                         const float* __restrict__ gamma, const float* __restrict__ beta) {
  long long tid = blockIdx.x * (long long)blockDim.x + threadIdx.x;
  const long long total  = (long long)n * d;
  const long long stride = (long long)gridDim.x * blockDim.x;
  for (; tid < total; tid += stride) {
    const int r = (int)(tid / d);
    const int c = (int)(tid % d);
    float* p = J + (size_t)r * LD + off + c;
    *p = gamma[c] * (*p - mean[c]) * inv[c] + beta[c];
  }
}

// ---------------------------------------------------------------- launcher
extern "C" void kernel_launch(void* const* d_in, const int* in_sizes, int n_in,
                              void* d_out, int out_size, void* d_ws, size_t ws_size,
                              hipStream_t stream) {
  const float* x    = (const float*)d_in[0];
  const int*   e1r  = (const int*)  d_in[1];
  const int*   e1c  = (const int*)  d_in[2];
  const float* e1v  = (const float*)d_in[3];
  const int*   e2r  = (const int*)  d_in[4];
  const int*   e2c  = (const int*)  d_in[5];
  const float* e2v  = (const float*)d_in[6];
  const float* We   = (const float*)d_in[7];
  const float* be   = (const float*)d_in[8];
  const float* g0   = (const float*)d_in[9];
  const float* bb0  = (const float*)d_in[10];
  const float* g1   = (const float*)d_in[11];
  const float* bb1  = (const float*)d_in[12];
  const float* Wh   = (const float*)d_in[13];
  const float* bh   = (const float*)d_in[14];

  const int n  = in_sizes[0] / 256;   // 50000
  const int E1 = in_sizes[1];         // 800000
  const int E2 = in_sizes[4];         // 1600000

  float* J     = (float*)d_ws;                     // [n, 960]
  float* sum   = J + (size_t)n * LD;               // 256 floats
  float* sumsq = sum + 256;                        // 256 floats
  float* mean  = sumsq + 256;                      // 256 floats
  float* inv   = mean + 256;                       // 256 floats

  const int rowTiles = (n + 15) >> 4;
  const int gemmBlocks = (rowTiles + 3) / 4;       // 4 waves (128 thr) per block

  // 1) zero J + stats (conv columns and BN sums are atomically accumulated)
  zero_f32<<<4096, 256, 0, stream>>>(J, (long long)n * LD + 1024);

  // 2) h = relu(x @ W_embed + b) -> J[:,0:64]
  embed_gemm_wmma<<<gemmBlocks, 128, 0, stream>>>(x, We, be, J, n);

  // 3) conv(h): d=64
  spmm_scatter<<<4096, 256, 0, stream>>>(e1r, e1c, e1v, E1, J + 0, J + 64,  16);
  spmm_scatter<<<4096, 256, 0, stream>>>(e2r, e2c, e2v, E2, J + 0, J + 128, 16);

  // 4) BN over cols [64,192)
  bn_partial<<<1024, 128, 0, stream>>>(J, n, 64, 128, sum, sumsq);
  bn_finalize<<<1, 256, 0, stream>>>(sum, sumsq, n, 128, mean, inv);
  bn_apply<<<4096, 256, 0, stream>>>(J, n, 64, 128, mean, inv, g0, bb0);

  // 5) conv(c0): d=128
  spmm_scatter<<<4096, 256, 0, stream>>>(e1r, e1c, e1v, E1, J + 64, J + 192, 32);
  spmm_scatter<<<4096, 256, 0, stream>>>(e2r, e2c, e2v, E2, J + 64, J + 320, 32);

  // 6) BN over cols [192,448) — reuse stats buffers, re-zero first
  zero_f32<<<8, 128, 0, stream>>>(sum, 512);
  bn_partial<<<1024, 256, 0, stream>>>(J, n, 192, 256, sum, sumsq);
  bn_finalize<<<1, 256, 0, stream>>>(sum, sumsq, n, 256, mean, inv);
  bn_apply<<<4096, 256, 0, stream>>>(J, n, 192, 256, mean, inv, g1, bb1);

  // 7) conv(c1): d=256
  spmm_scatter<<<4096, 256, 0, stream>>>(e1r, e1c, e1v, E1, J + 192, J + 448, 64);
  spmm_scatter<<<4096, 256, 0, stream>>>(e2r, e2c, e2v, E2, J + 192, J + 704, 64);

  // 8) out = J @ W_head + b_head
  head_gemm_wmma<<<gemmBlocks, 128, 0, stream>>>(J, Wh, bh, (float*)d_out, n);

  (void)n_in; (void)out_size; (void)ws_size;
}